// DenseGrid_43344809952174
// MI455X (gfx1250) — compile-verified
//
#include <hip/hip_runtime.h>

typedef float f4 __attribute__((ext_vector_type(4)));

#define FEAT 3

struct __align__(4) F3 { float a, b, c; };

template <int RES>
__device__ __forceinline__ void gather_add(const float* __restrict__ cb,
                                           float x, float y,
                                           float& r0, float& r1, float& r2) {
    const float s = (float)(RES - 1);
    // x,y in [0,1) -> x*s >= 0, so truncating cvt == floor (saves v_floor_f32)
    int xi = (int)(x * s);
    int yi = (int)(y * s);
    int base = (xi + yi * RES) * FEAT;
    // 12-byte gather; L2-resident (all codebooks total ~16MB << 192MB L2)
    F3 v = *reinterpret_cast<const F3*>(cb + base);
    r0 += v.a;
    r1 += v.b;
    r2 += v.c;
}

__device__ __forceinline__ void point_feature(const float* __restrict__ cb0,
                                              const float* __restrict__ cb1,
                                              const float* __restrict__ cb2,
                                              const float* __restrict__ cb3,
                                              float x, float y,
                                              float& r0, float& r1, float& r2) {
    r0 = 0.f; r1 = 0.f; r2 = 0.f;
    gather_add<128>(cb0, x, y, r0, r1, r2);
    gather_add<256>(cb1, x, y, r0, r1, r2);
    gather_add<512>(cb2, x, y, r0, r1, r2);
    gather_add<1024>(cb3, x, y, r0, r1, r2);
}

__global__ void __launch_bounds__(256)
dense_grid_gather(const float* __restrict__ pts,
                  const float* __restrict__ cb0,
                  const float* __restrict__ cb1,
                  const float* __restrict__ cb2,
                  const float* __restrict__ cb3,
                  float* __restrict__ out, int npts) {
    const int t  = blockIdx.x * blockDim.x + threadIdx.x;
    const int i0 = t * 4;

    if (i0 + 3 < npts) {
        const f4* p4 = (const f4*)pts;
        // gfx1250 stream-ahead prefetch of the pts stream (global_prefetch_b8)
        __builtin_prefetch((const char*)(p4 + (size_t)t * 2) + 8192, 0, 1);

        // 8 coords (4 points) via two NT b128 loads — read-once stream,
        // keep it out of L2 so the codebooks stay resident.
        f4 ab = __builtin_nontemporal_load(p4 + (size_t)t * 2);
        f4 cd = __builtin_nontemporal_load(p4 + (size_t)t * 2 + 1);

        float s00, s01, s02, s10, s11, s12, s20, s21, s22, s30, s31, s32;
        point_feature(cb0, cb1, cb2, cb3, ab.x, ab.y, s00, s01, s02);
        point_feature(cb0, cb1, cb2, cb3, ab.z, ab.w, s10, s11, s12);
        point_feature(cb0, cb1, cb2, cb3, cd.x, cd.y, s20, s21, s22);
        point_feature(cb0, cb1, cb2, cb3, cd.z, cd.w, s30, s31, s32);

        // 12 output floats = 48B contiguous, 16B aligned: three NT b128 stores
        f4* o4 = (f4*)(out + (size_t)i0 * 3);
        f4 o0 = {s00, s01, s02, s10};
        f4 o1 = {s11, s12, s20, s21};
        f4 o2 = {s22, s30, s31, s32};
        __builtin_nontemporal_store(o0, o4 + 0);
        __builtin_nontemporal_store(o1, o4 + 1);
        __builtin_nontemporal_store(o2, o4 + 2);
    } else {
        // scalar tail (never taken for N=4194304; kept for generality).
        // Keep it tiny: no unroll/vectorize so the hot path dominates I-cache.
#pragma clang loop unroll(disable) vectorize(disable)
        for (int i = i0; i < npts; ++i) {
            float x = pts[2 * i + 0];
            float y = pts[2 * i + 1];
            float r0, r1, r2;
            point_feature(cb0, cb1, cb2, cb3, x, y, r0, r1, r2);
            out[3 * i + 0] = r0;
            out[3 * i + 1] = r1;
            out[3 * i + 2] = r2;
        }
    }
}

extern "C" void kernel_launch(void* const* d_in, const int* in_sizes, int n_in,
                              void* d_out, int out_size, void* d_ws, size_t ws_size,
                              hipStream_t stream) {
    const float* pts = (const float*)d_in[0];
    const float* cb0 = (const float*)d_in[1];
    const float* cb1 = (const float*)d_in[2];
    const float* cb2 = (const float*)d_in[3];
    const float* cb3 = (const float*)d_in[4];
    float* out = (float*)d_out;

    const int npts     = in_sizes[0] / 2;          // pts is [N, 2] flat
    const int nthreads = (npts + 3) / 4;           // 4 points per thread
    const int block    = 256;                      // 8 wave32 waves
    const int grid     = (nthreads + block - 1) / block;

    dense_grid_gather<<<grid, block, 0, stream>>>(pts, cb0, cb1, cb2, cb3, out, npts);
}